// NN_12841952215599
// MI455X (gfx1250) — compile-verified
//
#include <hip/hip_runtime.h>
#include <hip/hip_bf16.h>

// out[B,10] = x[B,64] @ W[64,10], f32. HBM-bound (AI ~4.3 flop/B):
// floor = ~620MB / 23.3 TB/s ~= 27us. Strategy:
//   - x tile (128x64 f32, 32KB): GLOBAL_LOAD_ASYNC_TO_LDS_B128 (ASYNCcnt path,
//     no VGPR pass-through), padded LDS stride for conflict-free WMMA-layout reads
//   - W (2.5KB): staged to LDS once per block, B-fragments built via ds loads
//     (keeps the VMEM pipe free for the HBM x-stream)
//   - 16x V_WMMA_F32_16X16X4_F32 chain per wave (full f32 precision, K=64)

typedef __attribute__((ext_vector_type(2))) float v2f;
typedef __attribute__((ext_vector_type(4))) float v4f;
typedef __attribute__((ext_vector_type(8))) float v8f;
typedef __attribute__((ext_vector_type(4))) int   v4i;

typedef __attribute__((address_space(1))) v4i* gvp4;   // global v4i*
typedef __attribute__((address_space(3))) v4i* lvp4;   // LDS v4i*

#define ROWS_PER_BLOCK 128   // 8 waves x 16-row tiles
#define IN_DIM         64
#define OUT_DIM        10
#define LDS_STRIDE     68    // floats; 68 mod 64-banks = 4 -> 16 rows hit 64 distinct banks

#if defined(__has_builtin)
#  if __has_builtin(__builtin_amdgcn_global_load_async_to_lds_b128)
#    define USE_ASYNC_LDS 1
#  endif
#endif

__global__ __launch_bounds__(256) void neat_dense_wmma(
    const float* __restrict__ x,
    const float* __restrict__ W,
    float* __restrict__ out)
{
    __shared__ float lx[ROWS_PER_BLOCK * LDS_STRIDE];  // 34,816 B x-tile
    __shared__ float lw[IN_DIM * OUT_DIM];             // 2,560 B raw W copy

    const int tid  = threadIdx.x;
    const int lane = tid & 31;
    const int wave = tid >> 5;
    const int half = lane >> 4;   // 0: K pair {0,1}; 1: K pair {2,3} of each step
    const int col  = lane & 15;   // N index (output column) for B/D fragments
    const int m    = lane & 15;   // M index (row within 16-row tile) for A

    const float* xblk = x + (size_t)blockIdx.x * (ROWS_PER_BLOCK * IN_DIM);

    // ---- Stage x tile into LDS: 2048 v4f chunks, 8 per thread, coalesced ----
#if defined(USE_ASYNC_LDS)
    #pragma unroll
    for (int i = 0; i < 8; ++i) {
        const int idx = i * 256 + tid;   // v4f index within tile
        const int r   = idx >> 4;        // row 0..127
        const int c4  = idx & 15;        // 16B chunk within row
        __builtin_amdgcn_global_load_async_to_lds_b128(
            (gvp4)(xblk + r * IN_DIM + c4 * 4),
            (lvp4)(&lx[r * LDS_STRIDE + c4 * 4]),
            /*imm offset=*/0, /*cpol=*/0);
    }
#else
    #pragma unroll
    for (int i = 0; i < 8; ++i) {
        const int idx = i * 256 + tid;
        const int r   = idx >> 4;
        const int c4  = idx & 15;
        const v4f v = *(const v4f*)(xblk + r * IN_DIM + c4 * 4);
        *(v4f*)(&lx[r * LDS_STRIDE + c4 * 4]) = v;
    }
#endif

    // ---- Stage W (640 f32 = 160 v4f) into LDS, coalesced, once per block ----
    if (tid < (IN_DIM * OUT_DIM) / 4) {
        const v4f wv = *(const v4f*)(W + tid * 4);
        *(v4f*)(&lw[tid * 4]) = wv;
    }
    __syncthreads();   // W visible to all waves (x asyncs still in flight)

    // ---- Build B fragments from LDS: W padded 10->16 cols, ISA B-layout ----
    // Step s covers K = 4s..4s+3. Lane(col,half): v0=W[4s+2h][col], v1=W[4s+2h+1][col].
    v2f bfrag[16];
    #pragma unroll
    for (int s = 0; s < 16; ++s) {
        const int k0 = 4 * s + 2 * half;
        float b0 = 0.0f, b1 = 0.0f;
        if (col < OUT_DIM) {
            b0 = lw[k0 * OUT_DIM + col];
            b1 = lw[(k0 + 1) * OUT_DIM + col];
        }
        v2f bf; bf.x = b0; bf.y = b1;
        bfrag[s] = bf;
    }

#if defined(USE_ASYNC_LDS)
#  if __has_builtin(__builtin_amdgcn_s_wait_asynccnt)
    __builtin_amdgcn_s_wait_asynccnt(0);
#  else
    asm volatile("s_wait_asynccnt 0" ::: "memory");
#  endif
#endif
    __syncthreads();   // x tile visible to all waves

    // ---- 16x16 WMMA tile per wave: D = sum_s A(16x4) x B(4x16) ----
    // A-layout (ISA 7.12.2, 32-bit A 16x4): lanes 0-15 = row m, K={4s,4s+1};
    // lanes 16-31 = row m, K={4s+2,4s+3}  -> one b64 LDS read per step
    // (compiler pairs adjacent steps into ds_load_2addr_b64).
    const int arow = wave * 16 + m;
    v8f acc = {};
    #pragma unroll
    for (int s = 0; s < 16; ++s) {
        const int k0 = 4 * s + 2 * half;
        const v2f a = *(const v2f*)(&lx[arow * LDS_STRIDE + k0]);
        acc = __builtin_amdgcn_wmma_f32_16x16x4_f32(
            /*neg_a=*/false, a,
            /*neg_b=*/false, bfrag[s],
            /*c_mod=*/(short)0, acc,
            /*reuse_a=*/false, /*reuse_b=*/false);
    }

    // ---- Store D: VGPR r holds M = r + 8*half, N = col (cols 0..9 valid) ----
    if (col < OUT_DIM) {
        const size_t row0 = (size_t)blockIdx.x * ROWS_PER_BLOCK + wave * 16 + 8 * half;
        size_t obase = row0 * OUT_DIM + col;
        #pragma unroll
        for (int r = 0; r < 8; ++r) {
            out[obase + (size_t)r * OUT_DIM] = acc[r];
        }
    }
}

extern "C" void kernel_launch(void* const* d_in, const int* in_sizes, int n_in,
                              void* d_out, int out_size, void* d_ws, size_t ws_size,
                              hipStream_t stream) {
    const float* x = (const float*)d_in[0];   // [B, 64]
    const float* W = (const float*)d_in[1];   // [64, 10]
    float* out = (float*)d_out;               // [B, 10]

    const int batch  = in_sizes[0] / IN_DIM;          // 2,097,152
    const int blocks = batch / ROWS_PER_BLOCK;        // 16,384 (exact)

    neat_dense_wmma<<<blocks, 256, 0, stream>>>(x, W, out);
}